// InteractionGate_13683765805295
// MI455X (gfx1250) — compile-verified
//
#include <hip/hip_runtime.h>
#include <math.h>

typedef __attribute__((ext_vector_type(2))) float v2f;
typedef __attribute__((ext_vector_type(8))) float v8f;

#define HDIM 64
#define DDIM 32
#define FCIN 192   // 2*H + 2*D

// ---------------------------------------------------------------------------
// Setup: fold the rank-1 distance-embedding terms and the constant h[idx]
// block of fc_in into 64-vectors u1, u2, c0; also compute goal_out directly.
//   new_h[n] = sigmoid( W2 @ h[n] + d1[n]*u1 + d2[n]*u2 + c0 )
// ws layout (floats): [0..63]=u1  [64..127]=u2  [128..191]=c0  [192..255]=h[idx]
// ---------------------------------------------------------------------------
__global__ __launch_bounds__(64) void ig_setup(
    const float* __restrict__ h, const float* __restrict__ ghs,
    const float* __restrict__ goal, const float* __restrict__ position,
    const float* __restrict__ Wemb, const float* __restrict__ bemb,
    const float* __restrict__ Wfc, const float* __restrict__ bfc,
    const int* __restrict__ idxp, float* __restrict__ ws,
    float* __restrict__ goal_out)
{
    const int j   = threadIdx.x;      // 0..63 : output feature
    const int idx = idxp[0];
    const float* wrow = Wfc + (long)j * FCIN;

    float u1 = 0.f, u2 = 0.f, cb = 0.f;
    for (int t = 0; t < DDIM; ++t) {
        float wsum = Wemb[2*t] + Wemb[2*t+1];
        float wa = wrow[128 + t], wb = wrow[160 + t];
        u1 += wa * wsum;
        u2 += wb * wsum;
        cb += (wa + wb) * bemb[t];
    }
    float ch = 0.f, gsum = 0.f;
    for (int k = 0; k < HDIM; ++k) {
        ch   += wrow[k]      * h[(long)idx * HDIM + k];
        gsum += wrow[64 + k] * ghs[k];
    }
    float c0 = ch + cb + bfc[j];

    // goal update: dg = ||position[idx] - goal[idx]||
    float px = position[2*idx], py = position[2*idx + 1];
    float gx = goal[2*idx],     gy = goal[2*idx + 1];
    float dx = px - gx, dy = py - gy;
    float dg = sqrtf(dx*dx + dy*dy);
    float gv = ch + gsum + dg * (u1 + u2) + cb + bfc[j];
    goal_out[j] = 1.0f / (1.0f + __expf(-gv));

    ws[j]        = u1;
    ws[64 + j]   = u2;
    ws[128 + j]  = c0;
    ws[192 + j]  = h[(long)idx * HDIM + j];
}

// ---------------------------------------------------------------------------
// Main: (N x 64) @ (64 x 64)^T via V_WMMA_F32_16X16X4_F32.
// 8 waves / block, 16 agents / wave, 4 j-tile accumulators, 16 K-steps of 4.
// B (=W2 = W_fc[:,64:128]) is gathered into LDS pre-swizzled in exact
// per-instruction fragment order -> conflict-free ds_load_b64.
// ---------------------------------------------------------------------------
__global__ __launch_bounds__(256) void ig_main(
    const float* __restrict__ h, const float* __restrict__ goal,
    const float* __restrict__ position, const float* __restrict__ Wfc,
    const int* __restrict__ idxp, const float* __restrict__ ws,
    float* __restrict__ out, int N)
{
    __shared__ float sB[4096];   // 64 fragments * 32 lanes * 2 floats
    __shared__ float sE[256];    // u1 | u2 | c0 | h[idx]

    const int tid = threadIdx.x;

    // Cooperative gather of W2 into B-fragment layout:
    // fragment (jt,ks): lane holds B[k=4ks+2*(lane>>4)+c][n=(lane&15)]
    //                  = W2[jt*16+(lane&15)][4ks+2*(lane>>4)+c]
    for (int e = tid; e < 4096; e += 256) {
        int frag = e >> 6;
        int lane = (e >> 1) & 31;
        int c    = e & 1;
        int jt   = frag >> 4, ks = frag & 15;
        int j    = (jt << 4) + (lane & 15);
        int k    = (ks << 2) + ((lane >> 4) << 1) + c;
        sB[e] = Wfc[(long)j * FCIN + 64 + k];
    }
    sE[tid] = ws[tid];
    __syncthreads();

    const int lane = tid & 31;
    const int wave = tid >> 5;
    const long rowbase = ((long)blockIdx.x * 8 + wave) * 16;
    if (rowbase >= N) return;          // uniform per wave: EXEC stays all-ones

    const int idx = idxp[0];
    const float ax = position[2*idx];
    const float ay = position[2*idx + 1];

    const int  m    = lane & 15;       // M index (A layout: lanes 0-15 == 16-31)
    const int  hi   = lane >> 4;       // K half: 0 -> K{0,1}, 1 -> K{2,3}
    const int  koff = hi << 1;
    long row = rowbase + m;
    if (row >= N) row = N - 1;         // clamp (harmless duplicate loads)

    // A fragments: a[ks] = { h[row][4ks+koff], h[row][4ks+koff+1] }
    const float* hrow = h + row * HDIM;
    v2f a[16];
#pragma unroll
    for (int ks = 0; ks < 16; ++ks)
        a[ks] = *(const v2f*)(hrow + (ks << 2) + koff);

    // Faithful per-row distance math (a1=g1=position[idx], a2=goal, g2=position)
    float gx = goal[2*row],     gy = goal[2*row + 1];
    float px = position[2*row], py = position[2*row + 1];
    float dn      = -(ay - gy) * (ax - px);
    float cross_a = ax * gy - ay * gx;
    float cross_g = ax * py - ax * px;
    float n1 = cross_a * (ax - px) - (ax - gx) * cross_g;
    float n2 = cross_a * (ay - py) - (ay - gx) * cross_g;
    float sdn = (dn != 0.0f) ? dn : 1.0f;
    float p1 = n1 / sdn, p2 = n2 / sdn;
    bool cond = (dn != 0.0f) && ((ax - p1) * (p1 - ax) > 0.0f);
    float d2b = sqrtf((gx - px) * (gx - px) + (gy - py) * (gy - py));
    float d1v = cond ? sqrtf((ax - p1) * (ax - p1) + (ay - p2) * (ay - p2)) : 0.0f;
    float d2v = cond ? sqrtf((gx - p1) * (gx - p1) + (gy - p2) * (gy - p2)) : d2b;

    // GEMM: 4 j-tiles x 16 K-steps of f32 WMMA
    v8f acc[4] = {v8f{}, v8f{}, v8f{}, v8f{}};
    const v2f* sB2 = (const v2f*)sB;
#pragma unroll
    for (int ks = 0; ks < 16; ++ks) {
        v2f b0 = sB2[(0 * 16 + ks) * 32 + lane];
        v2f b1 = sB2[(1 * 16 + ks) * 32 + lane];
        v2f b2 = sB2[(2 * 16 + ks) * 32 + lane];
        v2f b3 = sB2[(3 * 16 + ks) * 32 + lane];
        acc[0] = __builtin_amdgcn_wmma_f32_16x16x4_f32(false, a[ks], false, b0, (short)0, acc[0], false, false);
        acc[1] = __builtin_amdgcn_wmma_f32_16x16x4_f32(false, a[ks], false, b1, (short)0, acc[1], false, false);
        acc[2] = __builtin_amdgcn_wmma_f32_16x16x4_f32(false, a[ks], false, b2, (short)0, acc[2], false, false);
        acc[3] = __builtin_amdgcn_wmma_f32_16x16x4_f32(false, a[ks], false, b3, (short)0, acc[3], false, false);
    }

    // Epilogue: rank-2 update + bias + sigmoid + row==idx passthrough
    float u1f[4], u2f[4], c0f[4], hif[4];
#pragma unroll
    for (int jt = 0; jt < 4; ++jt) {
        int j = (jt << 4) + m;
        u1f[jt] = sE[j];
        u2f[jt] = sE[64 + j];
        c0f[jt] = sE[128 + j];
        hif[jt] = sE[192 + j];
    }

#pragma unroll
    for (int r = 0; r < 8; ++r) {
        int src = r + (hi << 3);                 // C/D layout: vgpr r -> rows r, r+8
        float dd1 = __shfl(d1v, src, 32);
        float dd2 = __shfl(d2v, src, 32);
        long orow = rowbase + src;
        if (orow >= N) continue;
        float* orowp = out + orow * HDIM;
        bool keep = (orow == (long)idx);
#pragma unroll
        for (int jt = 0; jt < 4; ++jt) {
            float val = acc[jt][r] + dd1 * u1f[jt] + dd2 * u2f[jt] + c0f[jt];
            val = 1.0f / (1.0f + __expf(-val));
            if (keep) val = hif[jt];
            orowp[(jt << 4) + m] = val;
        }
    }
}

extern "C" void kernel_launch(void* const* d_in, const int* in_sizes, int n_in,
                              void* d_out, int out_size, void* d_ws, size_t ws_size,
                              hipStream_t stream)
{
    const float* h    = (const float*)d_in[0];   // (N, 64)
    const float* ghs  = (const float*)d_in[1];   // (64,)
    const float* goal = (const float*)d_in[2];   // (N, 2)
    const float* pos  = (const float*)d_in[3];   // (N, 2)
    const float* Wemb = (const float*)d_in[4];   // (32, 2)
    const float* bemb = (const float*)d_in[5];   // (32,)
    const float* Wfc  = (const float*)d_in[6];   // (64, 192)
    const float* bfc  = (const float*)d_in[7];   // (64,)
    const int*   idxp = (const int*)d_in[8];     // scalar
    float* out = (float*)d_out;                  // h_out (N*64) ++ goal_out (64)
    float* ws  = (float*)d_ws;

    const int N = in_sizes[0] / HDIM;

    ig_setup<<<1, 64, 0, stream>>>(h, ghs, goal, pos, Wemb, bemb, Wfc, bfc,
                                   idxp, ws, out + (long)N * HDIM);

    const int blocks = (N + 127) / 128;          // 8 waves * 16 rows per block
    ig_main<<<blocks, 256, 0, stream>>>(h, goal, pos, Wfc, idxp, ws, out, N);
}